// Distor_Value_47588237639989
// MI455X (gfx1250) — compile-verified
//
#include <hip/hip_runtime.h>
#include <hip/hip_bf16.h>

// CDNA5 / gfx1250 WMMA types
typedef float v2f __attribute__((ext_vector_type(2)));
typedef float v8f __attribute__((ext_vector_type(8)));

#if defined(__gfx1250__) && !__has_builtin(__builtin_amdgcn_wmma_f32_16x16x4_f32)
#error "expected __builtin_amdgcn_wmma_f32_16x16x4_f32 on gfx1250"
#endif

namespace {
constexpr int N_SAMPLES     = 128;
constexpr int PAD           = 18;   // LDS row stride (floats): conflict-free for all access phases
constexpr int TILES         = 8;    // 128 / 16
constexpr int RAYS_PER_WAVE = 16;
constexpr int WAVES_PER_BLK = 2;
constexpr int RAYS_PER_BLK  = RAYS_PER_WAVE * WAVES_PER_BLK; // 32
}

__global__ __launch_bounds__(64)
void Distor_Value_47588237639989_kernel(const float* __restrict__ t_inters,
                                        const float* __restrict__ weights,
                                        const float* __restrict__ t_near,
                                        const float* __restrict__ t_far,
                                        float* __restrict__ out) {
    // Per-wave LDS slices: W (weights), V (w*u), I (w^2 * ds) each [128][PAD]
    __shared__ float smem[WAVES_PER_BLK * 3 * N_SAMPLES * PAD];

    const int tid  = threadIdx.x;
    const int lane = tid & 31;
    const int wv   = tid >> 5;
    const int half = lane >> 4;   // 0: lanes 0-15, 1: lanes 16-31
    const int m    = lane & 15;

    float* Wl = smem + wv * 3 * N_SAMPLES * PAD;
    float* Vl = Wl + N_SAMPLES * PAD;
    float* Il = Vl + N_SAMPLES * PAD;

    const int rayBase = blockIdx.x * RAYS_PER_BLK + wv * RAYS_PER_WAVE;

    // -------- Phase 1: coalesced load + prep into LDS --------
    for (int rr = 0; rr < RAYS_PER_WAVE; ++rr) {
        const int b  = rayBase + rr;
        const float tn  = t_near[b];
        const float tf  = t_far[b];
        const float inv = 1.0f / (tf - tn);
        const float2* ti = reinterpret_cast<const float2*>(t_inters) + (size_t)b * N_SAMPLES;
        const float*  wp = weights + (size_t)b * N_SAMPLES;
        #pragma unroll
        for (int it = 0; it < 4; ++it) {
            const int i = it * 32 + lane;
            const float2 t = ti[i];                 // (t_i, t_{i+1}); shared boundary is exact
            const float  w = wp[i];
            const float s0 = (t.x - tn) * inv;
            const float s1 = (t.y - tn) * inv;
            const float u  = 0.5f * (s0 + s1);
            Wl[i * PAD + rr] = w;
            Vl[i * PAD + rr] = w * u;
            Il[i * PAD + rr] = w * w * (s1 - s0);
        }
    }
    __syncthreads();

    // -------- Phase 2: constant A operands (ISA half-wave K-split layout) --------
    // A 16x4 (MxK): lanes 0-15 hold K={0,1} in vgpr{0,1}; lanes 16-31 hold K={2,3}.
    v2f aL[4];        // slices of inclusive lower-triangular ones (L[m][k] = k<=m)
    #pragma unroll
    for (int c = 0; c < 4; ++c) {
        const int kb = 4 * c + 2 * half;
        aL[c].x = (kb     <= m) ? 1.0f : 0.0f;
        aL[c].y = (kb + 1 <= m) ? 1.0f : 0.0f;
    }
    v2f aOnes; aOnes.x = 1.0f; aOnes.y = 1.0f;   // slices of all-ones 16x16

    // -------- Phase 3: prefix-sum GEMM (P = Ltri @ W) + fused reductions --------
    v8f cacc = {};                     // ones @ (sum of previous W tiles): row-broadcast colsums
    float acc1 = 0.0f, acc2 = 0.0f, acc3 = 0.0f, acc4 = 0.0f;
    float totReg = 0.0f;

    for (int J = 0; J < TILES; ++J) {
        // B 4x16 (KxN): vgpr0 = rows {2*half+0}, vgpr1 = rows {2*half+1} of each 4-row slice
        v2f bs[4];
        #pragma unroll
        for (int c = 0; c < 4; ++c) {
            const int row = J * 16 + 4 * c + 2 * half;
            bs[c].x = Wl[row       * PAD + m];
            bs[c].y = Wl[(row + 1) * PAD + m];
        }

        // D_J = Ltri16 @ W_J + cacc   (inclusive prefix sums P for this tile)
        v8f d = cacc;
        #pragma unroll
        for (int c = 0; c < 4; ++c)
            d = __builtin_amdgcn_wmma_f32_16x16x4_f32(false, aL[c], false, bs[c],
                                                      (short)0, d, false, false);
        // cacc += ones16 @ W_J   (carry colsum broadcast to later tiles)
        #pragma unroll
        for (int c = 0; c < 4; ++c)
            cacc = __builtin_amdgcn_wmma_f32_16x16x4_f32(false, aOnes, false, bs[c],
                                                         (short)0, cacc, false, false);

        // Fused per-lane reductions in D layout: lane l, vgpr k -> (sample 16J+k+8*half, ray m)
        #pragma unroll
        for (int k = 0; k < 8; ++k) {
            const int sample = J * 16 + k + 8 * half;
            const int idx    = sample * PAD + m;
            const float P = d[k];
            const float v = Vl[idx];
            const float w = Wl[idx];
            acc1 = fmaf(v, P, acc1);   // sum v_i * P_i
            acc2 = fmaf(v, w, acc2);   // sum v_i * w_i
            acc3 += v;                 // sum v_i
            acc4 += Il[idx];           // sum w_i^2 * ds_i
        }
        if (J == TILES - 1) totReg = d[7];   // half==1 lanes: P[127] = total weight per ray
    }

    // Fold the two half-wave sample partitions (lane l <-> l+16, same ray)
    acc1 += __shfl_xor(acc1, 16);
    acc2 += __shfl_xor(acc2, 16);
    acc3 += __shfl_xor(acc3, 16);
    acc4 += __shfl_xor(acc4, 16);
    const float other = __shfl_xor(totReg, 16);
    const float tot   = half ? totReg : other;

    if (lane < 16) {
        // loss_inter = 2 * sum_i v_i * (2*P_i - w_i - tot)   (sorted u, antisymmetric sign matrix)
        const float inter = 4.0f * acc1 - 2.0f * acc2 - 2.0f * tot * acc3;
        out[rayBase + lane] = inter + acc4 * (1.0f / 3.0f);
    }
}

extern "C" void kernel_launch(void* const* d_in, const int* in_sizes, int n_in,
                              void* d_out, int out_size, void* d_ws, size_t ws_size,
                              hipStream_t stream) {
    const float* t_inters = (const float*)d_in[0];   // [B, N, 2]
    const float* weights  = (const float*)d_in[1];   // [B, N]
    const float* t_near   = (const float*)d_in[2];   // [B, 1]
    const float* t_far    = (const float*)d_in[3];   // [B, 1]
    float* out = (float*)d_out;                      // [B]

    const int B = in_sizes[1] / N_SAMPLES;           // 4096
    dim3 grid(B / RAYS_PER_BLK);                     // 128 blocks
    dim3 block(WAVES_PER_BLK * 32);                  // 64 threads = 2 waves

    Distor_Value_47588237639989_kernel<<<grid, block, 0, stream>>>(
        t_inters, weights, t_near, t_far, out);
}